// ElevationLoss_635655160394
// MI455X (gfx1250) — compile-verified
//
#include <hip/hip_runtime.h>
#include <math.h>

// Problem constants from the reference: B=8, C=2, H=W=512.
#define HDIM 512
#define WDIM 512
#define HWSZ (HDIM * WDIM)      // 2^18
#define ROW_SHIFT 9             // W = 512
#define GPR 128                 // groups (of 4 px) per row
#define GHW (HWSZ / 4)          // groups per image = 65536
#define GHW_SHIFT 16
#define NT 256                  // 8 wave32s per block
#define NWAVE (NT / 32)
#define NACC 10                 // inter0,inter1,fp0,fp1,fn0,fn1,cnt0,cnt1,loss,comb
#define MAXBLOCKS 2048

typedef float v2f __attribute__((ext_vector_type(2)));
typedef float v4f __attribute__((ext_vector_type(4)));
typedef float v8f __attribute__((ext_vector_type(8)));
typedef int   v4i __attribute__((ext_vector_type(4)));

// Exact fp32 sum across all 32 lanes of a wave using V_WMMA_F32_16X16X4_F32.
// A(16x4 f32): lane L<16 supplies A[L][0]/A[L][1]; lanes 16-31 supply
// A[L-16][2]/A[L-16][3]. v in VGPR0, 0 in VGPR1, B = all ones, C = 0 =>
// D[m][n] = v[m] + v[m+16] replicated over columns. D layout: VGPR j holds
// row j (lanes 0-15) and row j+8 (lanes 16-31); summing the 8 result VGPRs
// gives rows 0-7 / 8-15 per half-wave, one xor-16 shuffle finishes the total.
// EXEC must be all ones at the call site (callers keep control flow uniform).
__device__ __forceinline__ float wave_sum32(float v) {
  v2f a; a[0] = v;    a[1] = 0.0f;
  v2f b; b[0] = 1.0f; b[1] = 1.0f;
  v8f c = {0.f, 0.f, 0.f, 0.f, 0.f, 0.f, 0.f, 0.f};
  c = __builtin_amdgcn_wmma_f32_16x16x4_f32(false, a, false, b, (short)0, c,
                                            false, false);
  float s = ((c[0] + c[1]) + (c[2] + c[3])) + ((c[4] + c[5]) + (c[6] + c[7]));
  s += __shfl_xor(s, 16, 32);
  return s;
}

__device__ __forceinline__ void block_reduce_store(float* acc, float* dst) {
  __shared__ float lds[NWAVE][NACC];
  const int lane = threadIdx.x & 31;
  const int wv = threadIdx.x >> 5;
#pragma unroll
  for (int k = 0; k < NACC; ++k) {
    float s = wave_sum32(acc[k]);   // uniform path: EXEC all ones
    if (lane == 0) lds[wv][k] = s;
  }
  __syncthreads();
  if (threadIdx.x == 0) {
#pragma unroll
    for (int k = 0; k < NACC; ++k) {
      float t = 0.f;
#pragma unroll
      for (int w = 0; w < NWAVE; ++w) t += lds[w][k];
      dst[k] = t;
    }
  }
}

// Load one stencil row (6 heights + 6 gt) for a 4-pixel group with zero
// padding: middle 4 elements as b128, the two edges as guarded b32.
__device__ __forceinline__ void load_row(const float* __restrict__ heights,
                                         const int* __restrict__ gt,
                                         size_t sb, int yy, int x4,
                                         float* h, int* g) {
  const bool rin = ((unsigned)yy < (unsigned)HDIM);
  if (rin) {
    const size_t ridx = sb + ((size_t)yy << ROW_SHIFT) + (size_t)x4;
    const v4f hm = *(const v4f*)(heights + ridx);
    const v4i gm = *(const v4i*)(gt + ridx);
    h[1] = hm[0]; h[2] = hm[1]; h[3] = hm[2]; h[4] = hm[3];
    g[1] = gm[0]; g[2] = gm[1]; g[3] = gm[2]; g[4] = gm[3];
    const bool lin = (x4 > 0);
    const bool rgt = (x4 + 4 < WDIM);
    h[0] = lin ? heights[ridx - 1] : 0.f;
    g[0] = lin ? gt[ridx - 1] : 0;
    h[5] = rgt ? heights[ridx + 4] : 0.f;
    g[5] = rgt ? gt[ridx + 4] : 0;
  } else {
#pragma unroll
    for (int k = 0; k < 6; ++k) { h[k] = 0.f; g[k] = 0; }
  }
}

// Pass 1: fused softmax + Tversky partials + 3x3 elevation stencil partials.
// Each thread owns 4 consecutive x-pixels (vectorized b128 loads).
__global__ void __launch_bounds__(NT) elev_partials(
    const float* __restrict__ pred,     // (B,2,H,W)
    const float* __restrict__ heights,  // (B,1,H,W)
    const int* __restrict__ gt,         // (B,1,H,W)
    float* __restrict__ ws, int nBatch) {
  const int Ng = nBatch * GHW;          // 4-pixel groups
  float acc[NACC];
#pragma unroll
  for (int k = 0; k < NACC; ++k) acc[k] = 0.f;

  const int stride = gridDim.x * blockDim.x;
  for (int gi = blockIdx.x * blockDim.x + threadIdx.x; gi < Ng; gi += stride) {
    const int b = gi >> GHW_SHIFT;
    const int rem = gi & (GHW - 1);
    const int y = rem >> 7;             // 128 groups per row
    const int x4 = (rem & (GPR - 1)) << 2;
    const size_t sb = (size_t)b * HWSZ;
    const size_t rowc = sb + ((size_t)y << ROW_SHIFT) + (size_t)x4;

    // logits: two b128 loads (class planes), no reuse
    const size_t pbase = (size_t)b * 2 * HWSZ + ((size_t)y << ROW_SHIFT) +
                         (size_t)x4;
    const v4f l0 = *(const v4f*)(pred + pbase);
    const v4f l1 = *(const v4f*)(pred + pbase + HWSZ);

    // stencil rows y-1, y, y+1: heights + gt, zero-padded
    float h0[6], h1[6], h2[6];
    int g0[6], g1[6], g2[6];
    load_row(heights, gt, sb, y - 1, x4, h0, g0);
    load_row(heights, gt, sb, y,     x4, h1, g1);
    load_row(heights, gt, sb, y + 1, x4, h2, g2);

#pragma unroll
    for (int px = 0; px < 4; ++px) {
      // 2-class softmax (== exp(log_softmax))
      const float a0 = l0[px];
      const float a1 = l1[px];
      const float m = fmaxf(a0, a1);
      const float e0 = expf(a0 - m);
      const float e1 = expf(a1 - m);
      const float inv = 1.0f / (e0 + e1);
      const float p0 = e0 * inv;
      const float p1 = e1 * inv;

      // Tversky partials (ignore_index=255: onehot zeroed, pred still in fp)
      const int g = g1[1 + px];         // center gt
      const bool valid = (g != 255);
      const float oh0 = (valid && g == 0) ? 1.f : 0.f;
      const float oh1 = (valid && g == 1) ? 1.f : 0.f;
      acc[0] += p0 * oh0;          // inter0
      acc[1] += p1 * oh1;          // inter1
      acc[2] += p0 * (1.f - oh0);  // fp0
      acc[3] += p1 * (1.f - oh1);  // fp1
      acc[4] += (1.f - p0) * oh0;  // fn0
      acc[5] += (1.f - p1) * oh1;  // fn1
      acc[6] += oh0;               // cnt0 (class presence)
      acc[7] += oh1;               // cnt1

      // unified_pred: argmax ties -> class 0 (jnp.argmax first-max semantics)
      const float u = (p1 > p0) ? p1 : -p0;
      const float hc = h1[1 + px];

      float lsum = 0.f, csum = 0.f;
      const float* hr[3] = {h0, h1, h2};
      const int* gr[3] = {g0, g1, g2};
#pragma unroll
      for (int r = 0; r < 3; ++r) {
#pragma unroll
        for (int dx = 0; dx < 3; ++dx) {
          const float hn = hr[r][px + dx];
          const int gn = gr[r][px + dx];
          const float gf = (gn == 1) ? 1.f : 0.f;    // gt_flood
          const float gd = (gn == 0) ? 1.f : 0.f;    // gt_dry (pad counts)
          const float ig = (gn != 255) ? 1.f : 0.f;  // ignore mask
          const float d = hc - hn;
          const float pe = (d > 0.f) ? 1.f : 0.f;
          const float ne = (d < 0.f) ? 1.f : 0.f;
          const float comb = ig * (1.f - gf * pe) * (1.f - gd * ne);
          const float score = 1.f - (gf - gd) * u;
          lsum += fmaxf(comb * score, 0.f);
          csum += comb;
        }
      }
      acc[8] += lsum;
      acc[9] += csum;
    }
    (void)rowc;
  }

  block_reduce_store(acc, ws + (size_t)blockIdx.x * NACC);
}

// Pass 2: deterministic reduction of block partials + closed-form loss.
__global__ void __launch_bounds__(NT) elev_final(
    const float* __restrict__ ws, int nblocks, float* __restrict__ out) {
  float acc[NACC];
#pragma unroll
  for (int k = 0; k < NACC; ++k) acc[k] = 0.f;
  for (int j = threadIdx.x; j < nblocks; j += blockDim.x) {
#pragma unroll
    for (int k = 0; k < NACC; ++k) acc[k] += ws[(size_t)j * NACC + k];
  }

  __shared__ float tot[NACC];
  block_reduce_store(acc, tot);
  __syncthreads();

  if (threadIdx.x == 0) {
    // Tversky: mean over C=2 classes of present * (1 - inter/denom)^gamma
    float tversky = 0.f;
#pragma unroll
    for (int c = 0; c < 2; ++c) {
      const float inter = tot[0 + c];
      const float fp = tot[2 + c];
      const float fn = tot[4 + c];
      const float cnt = tot[6 + c];
      const float denom =
          fmaxf(inter + 0.3f * fp + 0.7f * fn + 1e-7f, 1e-7f);
      const float score = inter / denom;
      const float lossc = powf(1.f - score, 1.33f);
      const float present = (cnt > 0.f) ? 1.f : 0.f;
      tversky += lossc * present;
    }
    tversky *= 0.5f;

    const float n_valid = fmaxf(tot[9], 1.f);
    const float loss_elev = tot[8] / n_valid;
    out[0] = tversky * loss_elev;
  }
}

extern "C" void kernel_launch(void* const* d_in, const int* in_sizes, int n_in,
                              void* d_out, int out_size, void* d_ws,
                              size_t ws_size, hipStream_t stream) {
  const float* pred = (const float*)d_in[0];     // pred_labels (B,2,H,W) f32
  const float* heights = (const float*)d_in[1];  // heights (B,1,H,W) f32
  const int* gt = (const int*)d_in[2];           // gt_labels (B,1,H,W) i32
  float* out = (float*)d_out;
  float* ws = (float*)d_ws;

  const int nBatch = in_sizes[1] / HWSZ;  // heights has B*H*W elements

  int nb = (int)(ws_size / (NACC * sizeof(float)));
  if (nb > MAXBLOCKS) nb = MAXBLOCKS;
  if (nb < 1) nb = 1;

  elev_partials<<<nb, NT, 0, stream>>>(pred, heights, gt, ws, nBatch);
  elev_final<<<1, NT, 0, stream>>>(ws, nb, out);
}